// EMDLoss_19164144075467
// MI455X (gfx1250) — compile-verified
//
#include <hip/hip_runtime.h>

// ---------------------------------------------------------------------------
// EMD (approxmatch) for B=8, N=M=4096 on MI455X (gfx1250).
//  - d2 recomputed from raw points each pass (points = 48KB/batch; materialized
//    d2 would be 537MB -> ~16GB HBM traffic over 30 passes).
//  - x·y Gram tiles via V_WMMA_F32_16X16X4_F32 (K=3 padded to 4).
//  - The streamed side (all 4096 opposite points + folded scale arrays) is
//    staged once per block into LDS (~80KB of the WGP's 320KB), so the 256-tile
//    sweep reads broadcast ds_loads instead of re-hitting VMEM per wave.
//  - 'level' is folded into the staged constants so the pre-exp VALU chain is
//    just add + fma before v_exp_f32 (TRANS co-executes with VALU).
// ---------------------------------------------------------------------------

#define B_ 8
#define N_ 4096
#define M_ 4096
#define WAVES_PER_BLOCK 8
#define THREADS_PER_BLOCK 256
#define ROW_TILES (N_ / 16)                              // 256 tiles per batch
#define COL_TILES (M_ / 16)
#define GRID_ROWS (B_ * ROW_TILES / WAVES_PER_BLOCK)     // 256 blocks
#define GRID_COLS (B_ * COL_TILES / WAVES_PER_BLOCK)     // 256 blocks

// LDS staging: 3*4096 point floats + 2 * 4096 scalar arrays = 81920 bytes
#define SMEM_PTS   (3 * M_)
#define SMEM_FLOATS (SMEM_PTS + 2 * M_)
#define SMEM_BYTES  (SMEM_FLOATS * 4)

typedef float v2f __attribute__((ext_vector_type(2)));
typedef float v8f __attribute__((ext_vector_type(8)));

// D = A(16x4 f32) * B(4x16 f32) + 0  -> 16x16 f32 across the wave.
__device__ __forceinline__ v8f wmma_dot(v2f a, v2f b) {
  v8f c = {};
  return __builtin_amdgcn_wmma_f32_16x16x4_f32(
      /*neg_a=*/false, a, /*neg_b=*/false, b,
      /*c_mod=*/(short)0, c, /*reuse_a=*/false, /*reuse_b=*/false);
}

// K-striped A/B operand: lanes 0-15 -> (K0,K1)=(c0,c1); lanes 16-31 -> (K2,0).
__device__ __forceinline__ v2f make_op(const float* __restrict__ p, int half) {
  v2f r;
  r.x = half ? p[2] : p[0];
  r.y = half ? 0.0f : p[1];
  return r;
}

// ---------------------------------------------------------------------------
__global__ void emd_init(const float* __restrict__ x, const float* __restrict__ y,
                         float* __restrict__ satl, float* __restrict__ satr,
                         float* __restrict__ sqn1, float* __restrict__ sqn2,
                         float* __restrict__ partial) {
  const int i = blockIdx.x * blockDim.x + threadIdx.x;
  const float mx = (float)(N_ > M_ ? N_ : M_);
  if (i < B_ * N_) {
    const float* p = x + (size_t)i * 3;
    sqn1[i] = p[0] * p[0] + p[1] * p[1] + p[2] * p[2];
    satl[i] = mx / (float)N_;
  }
  if (i < B_ * M_) {
    const float* p = y + (size_t)i * 3;
    sqn2[i] = p[0] * p[0] + p[1] * p[1] + p[2] * p[2];
    satr[i] = mx / (float)M_;
  }
  if (i < GRID_ROWS) partial[i] = 0.0f;
}

// ---------------------------------------------------------------------------
// pass 1: rowsum[n] = sum_m exp(level*d2[n,m]) * satr[m]
// LDS: y points, level*sqn2, satr.  exp arg = (l*s1 + l*s2) + (-2*level)*dot.
// ---------------------------------------------------------------------------
__global__ void __launch_bounds__(THREADS_PER_BLOCK)
emd_rowsum(const float* __restrict__ x, const float* __restrict__ y,
           const float* __restrict__ sqn1, const float* __restrict__ sqn2,
           const float* __restrict__ satr, float* __restrict__ rowsum, float level) {
  extern __shared__ float smem[];
  float* sY  = smem;                 // [3*M_]
  float* sL2 = smem + SMEM_PTS;      // level * sqn2
  float* sSr = sL2 + M_;             // satr

  const int lane = threadIdx.x & 31;
  const int wave = threadIdx.x >> 5;
  const int half = lane >> 4;
  const int l = lane & 15;
  const int gw = blockIdx.x * WAVES_PER_BLOCK + wave;
  const int b = gw / ROW_TILES;
  const int rowBase = (gw % ROW_TILES) * 16;

  const float* xb = x + (size_t)b * N_ * 3;
  const float* yb = y + (size_t)b * M_ * 3;
  const float* sq1b = sqn1 + (size_t)b * N_;
  const float* sq2b = sqn2 + (size_t)b * M_;
  const float* satrb = satr + (size_t)b * M_;

  for (int i = threadIdx.x; i < SMEM_PTS; i += THREADS_PER_BLOCK) sY[i] = yb[i];
  for (int i = threadIdx.x; i < M_; i += THREADS_PER_BLOCK) {
    sL2[i] = level * sq2b[i];        // fold level during the copy
    sSr[i] = satrb[i];
  }
  __syncthreads();

  const v2f a = make_op(xb + (size_t)(rowBase + l) * 3, half);
  const float m2l = -2.0f * level;

  float ls1[8], acc[8];
#pragma unroll
  for (int v = 0; v < 8; ++v) {
    ls1[v] = level * sq1b[rowBase + v + 8 * half];  // C layout: row v + 8*half
    acc[v] = 0.0f;
  }

  for (int ct = 0; ct < COL_TILES; ++ct) {
    const int col = ct * 16 + l;                    // C layout: col = lane & 15
    const v2f bt = make_op(sY + col * 3, half);
    const float ls2 = sL2[col];
    const float sr = sSr[col];
    const v8f c = wmma_dot(a, bt);
#pragma unroll
    for (int v = 0; v < 8; ++v) {
      const float e = fmaf(m2l, c[v], ls1[v] + ls2);
      acc[v] = fmaf(__expf(e), sr, acc[v]);
    }
  }

#pragma unroll
  for (int v = 0; v < 8; ++v) {
    float t = acc[v];
    t += __shfl_xor(t, 1);
    t += __shfl_xor(t, 2);
    t += __shfl_xor(t, 4);
    t += __shfl_xor(t, 8);
    if (l == 0) rowsum[(size_t)b * N_ + rowBase + v + 8 * half] = t;
  }
}

// coef[n] = satl[n] / (rowsum[n] + 1e-9)
__global__ void emd_coef(const float* __restrict__ satl, const float* __restrict__ rowsum,
                         float* __restrict__ coef) {
  const int i = blockIdx.x * blockDim.x + threadIdx.x;
  if (i < B_ * N_) coef[i] = satl[i] / (rowsum[i] + 1e-9f);
}

// ---------------------------------------------------------------------------
// pass 2: ss2[m] = satr[m] * sum_n exp(level*d2[n,m]) * coef[n]
//         ratio[m] = min(satr[m]/(ss2[m]+1e-9), 1)
// LDS: x points, level*sqn1, coef.  B operand (columns) hoisted per wave.
// ---------------------------------------------------------------------------
__global__ void __launch_bounds__(THREADS_PER_BLOCK)
emd_colsum(const float* __restrict__ x, const float* __restrict__ y,
           const float* __restrict__ sqn1, const float* __restrict__ sqn2,
           const float* __restrict__ satr, const float* __restrict__ coef,
           float* __restrict__ ss2, float* __restrict__ ratio, float level) {
  extern __shared__ float smem[];
  float* sX  = smem;                 // [3*N_]
  float* sL1 = smem + SMEM_PTS;      // level * sqn1
  float* sCf = sL1 + N_;             // coef

  const int lane = threadIdx.x & 31;
  const int wave = threadIdx.x >> 5;
  const int half = lane >> 4;
  const int l = lane & 15;
  const int gw = blockIdx.x * WAVES_PER_BLOCK + wave;
  const int b = gw / COL_TILES;
  const int colBase = (gw % COL_TILES) * 16;

  const float* xb = x + (size_t)b * N_ * 3;
  const float* yb = y + (size_t)b * M_ * 3;
  const float* sq1b = sqn1 + (size_t)b * N_;
  const float* sq2b = sqn2 + (size_t)b * M_;
  const float* coefb = coef + (size_t)b * N_;

  for (int i = threadIdx.x; i < SMEM_PTS; i += THREADS_PER_BLOCK) sX[i] = xb[i];
  for (int i = threadIdx.x; i < N_; i += THREADS_PER_BLOCK) {
    sL1[i] = level * sq1b[i];
    sCf[i] = coefb[i];
  }
  __syncthreads();

  const v2f bt = make_op(yb + (size_t)(colBase + l) * 3, half);
  const int col = colBase + l;
  const float ls2 = level * sq2b[col];
  const float sr = satr[(size_t)b * M_ + col];
  const float m2l = -2.0f * level;

  float acc = 0.0f;
  for (int rt = 0; rt < ROW_TILES; ++rt) {
    const int rowBase = rt * 16;
    const v2f a = make_op(sX + (rowBase + l) * 3, half);
    const v8f c = wmma_dot(a, bt);
#pragma unroll
    for (int v = 0; v < 8; ++v) {
      const int row = rowBase + v + 8 * half;
      const float e = fmaf(m2l, c[v], sL1[row] + ls2);   // broadcast ds reads
      acc = fmaf(__expf(e), sCf[row], acc);
    }
  }
  acc *= sr;
  acc += __shfl_xor(acc, 16);                            // join the two halves
  if (half == 0) {
    const size_t i = (size_t)b * M_ + col;
    ss2[i] = acc;
    ratio[i] = fminf(sr / (acc + 1e-9f), 1.0f);
  }
}

// ---------------------------------------------------------------------------
// pass 3: w' = exp(level*d2)*satr[m]*ratio[m]*coef[n]
//         cost += w'*d2 ; satl[n] = max(satl[n] - sum_m w', 0)
// LDS: y points, sqn2 (raw, needed for d2), satr*ratio (folded).
// ---------------------------------------------------------------------------
__global__ void __launch_bounds__(THREADS_PER_BLOCK)
emd_update(const float* __restrict__ x, const float* __restrict__ y,
           const float* __restrict__ sqn1, const float* __restrict__ sqn2,
           const float* __restrict__ satr, const float* __restrict__ coef,
           const float* __restrict__ ratio, float* __restrict__ satl,
           float* __restrict__ partial, float level) {
  extern __shared__ float smem[];
  float* sY   = smem;                // [3*M_]
  float* sS2  = smem + SMEM_PTS;     // sqn2 (raw)
  float* sSrr = sS2 + M_;            // satr * ratio

  const int lane = threadIdx.x & 31;
  const int wave = threadIdx.x >> 5;
  const int half = lane >> 4;
  const int l = lane & 15;
  const int gw = blockIdx.x * WAVES_PER_BLOCK + wave;
  const int b = gw / ROW_TILES;
  const int rowBase = (gw % ROW_TILES) * 16;

  const float* xb = x + (size_t)b * N_ * 3;
  const float* yb = y + (size_t)b * M_ * 3;
  const float* sq1b = sqn1 + (size_t)b * N_;
  const float* sq2b = sqn2 + (size_t)b * M_;
  const float* satrb = satr + (size_t)b * M_;
  const float* ratb = ratio + (size_t)b * M_;
  const float* coefb = coef + (size_t)b * N_;

  for (int i = threadIdx.x; i < SMEM_PTS; i += THREADS_PER_BLOCK) sY[i] = yb[i];
  for (int i = threadIdx.x; i < M_; i += THREADS_PER_BLOCK) {
    sS2[i] = sq2b[i];
    sSrr[i] = satrb[i] * ratb[i];    // fold satr*ratio during the copy
  }
  __syncthreads();

  const v2f a = make_op(xb + (size_t)(rowBase + l) * 3, half);

  float s1[8], cf[8], accw[8];
#pragma unroll
  for (int v = 0; v < 8; ++v) {
    const int row = rowBase + v + 8 * half;
    s1[v] = sq1b[row];
    cf[v] = coefb[row];
    accw[v] = 0.0f;
  }
  float accc = 0.0f;

  for (int ct = 0; ct < COL_TILES; ++ct) {
    const int col = ct * 16 + l;
    const v2f bt = make_op(sY + col * 3, half);
    const float s2 = sS2[col];
    const float srr = sSrr[col];
    const v8f c = wmma_dot(a, bt);
#pragma unroll
    for (int v = 0; v < 8; ++v) {
      const float d2 = fmaf(-2.0f, c[v], s1[v] + s2);
      const float w = __expf(level * d2) * srr * cf[v];
      accw[v] += w;
      accc = fmaf(w, d2, accc);
    }
  }

#pragma unroll
  for (int v = 0; v < 8; ++v) {      // row marginals -> satl update
    float t = accw[v];
    t += __shfl_xor(t, 1);
    t += __shfl_xor(t, 2);
    t += __shfl_xor(t, 4);
    t += __shfl_xor(t, 8);
    if (l == 0) {
      const size_t i = (size_t)b * N_ + rowBase + v + 8 * half;
      satl[i] = fmaxf(satl[i] - t, 0.0f);
    }
  }

  // deterministic cost reduction: wave -> LDS -> single writer per block slot
  accc += __shfl_xor(accc, 1);
  accc += __shfl_xor(accc, 2);
  accc += __shfl_xor(accc, 4);
  accc += __shfl_xor(accc, 8);
  accc += __shfl_xor(accc, 16);
  __shared__ float wpart[WAVES_PER_BLOCK];
  if (lane == 0) wpart[wave] = accc;
  __syncthreads();
  if (threadIdx.x == 0) {
    float s = 0.0f;
#pragma unroll
    for (int w = 0; w < WAVES_PER_BLOCK; ++w) s += wpart[w];
    partial[blockIdx.x] += s;        // accumulates across the 10 levels
  }
}

// satr[m] = max(satr[m] - ss2[m]*ratio[m], 0)
__global__ void emd_satr_update(float* __restrict__ satr, const float* __restrict__ ss2,
                                const float* __restrict__ ratio) {
  const int i = blockIdx.x * blockDim.x + threadIdx.x;
  if (i < B_ * M_) satr[i] = fmaxf(satr[i] - ss2[i] * ratio[i], 0.0f);
}

__global__ void emd_finalize(const float* __restrict__ partial, float* __restrict__ out) {
  if (blockIdx.x == 0 && threadIdx.x == 0) {
    float s = 0.0f;
    for (int i = 0; i < GRID_ROWS; ++i) s += partial[i];
    out[0] = s / (float)((size_t)B_ * N_);  // mean over batch, then / N
  }
}

// ---------------------------------------------------------------------------
extern "C" void kernel_launch(void* const* d_in, const int* in_sizes, int n_in,
                              void* d_out, int out_size, void* d_ws, size_t ws_size,
                              hipStream_t stream) {
  (void)in_sizes; (void)n_in; (void)out_size; (void)ws_size;
  const float* x = (const float*)d_in[0];   // [B,N,3]
  const float* y = (const float*)d_in[1];   // [B,M,3]
  float* out = (float*)d_out;

  float* ws = (float*)d_ws;
  const int NN = B_ * N_, MM = B_ * M_;
  float* satl    = ws;            ws += NN;
  float* satr    = ws;            ws += MM;
  float* rowsum  = ws;            ws += NN;
  float* coef    = ws;            ws += NN;
  float* ss2     = ws;            ws += MM;
  float* ratio   = ws;            ws += MM;
  float* sqn1    = ws;            ws += NN;
  float* sqn2    = ws;            ws += MM;
  float* partial = ws;            ws += GRID_ROWS;

  const int initN = (NN > MM ? NN : MM);
  emd_init<<<(initN + THREADS_PER_BLOCK - 1) / THREADS_PER_BLOCK, THREADS_PER_BLOCK, 0, stream>>>(
      x, y, satl, satr, sqn1, sqn2, partial);

  // levels: -(4^7) .. -(4^-1), then 0 (exactly as approxmatch.cu)
  const float levels[10] = {-16384.0f, -4096.0f, -1024.0f, -256.0f, -64.0f,
                            -16.0f, -4.0f, -1.0f, -0.25f, 0.0f};
  const int egN = (NN + THREADS_PER_BLOCK - 1) / THREADS_PER_BLOCK;
  const int egM = (MM + THREADS_PER_BLOCK - 1) / THREADS_PER_BLOCK;

  for (int lv = 0; lv < 10; ++lv) {
    const float L = levels[lv];
    emd_rowsum<<<GRID_ROWS, THREADS_PER_BLOCK, SMEM_BYTES, stream>>>(
        x, y, sqn1, sqn2, satr, rowsum, L);
    emd_coef<<<egN, THREADS_PER_BLOCK, 0, stream>>>(satl, rowsum, coef);
    emd_colsum<<<GRID_COLS, THREADS_PER_BLOCK, SMEM_BYTES, stream>>>(
        x, y, sqn1, sqn2, satr, coef, ss2, ratio, L);
    emd_update<<<GRID_ROWS, THREADS_PER_BLOCK, SMEM_BYTES, stream>>>(
        x, y, sqn1, sqn2, satr, coef, ratio, satl, partial, L);
    emd_satr_update<<<egM, THREADS_PER_BLOCK, 0, stream>>>(satr, ss2, ratio);
  }

  emd_finalize<<<1, 32, 0, stream>>>(partial, out);
}